// VLLMKVCache_33002528702775
// MI455X (gfx1250) — compile-verified
//
#include <hip/hip_runtime.h>

// Problem geometry (matches the reference file).
#define NUM_BLOCKS_   1024
#define BLOCK_SIZE_   128
#define NUM_KV_HEADS_ 8
#define HEAD_DIM_     128
#define NUM_TOKENS_   16384

#define SLOT_ELEMS (NUM_KV_HEADS_ * HEAD_DIM_)   // 1024 floats per (block,offset) slot
#define SLOT_BYTES (SLOT_ELEMS * 4)              // 4096 bytes per slot
#define SLOT_VEC4  (SLOT_ELEMS / 4)              // 256 x 16B vectors per slot
#define NUM_SLOTS  (NUM_BLOCKS_ * BLOCK_SIZE_)   // 131072 slots total

typedef __attribute__((ext_vector_type(4))) float v4f;

// ---------------------------------------------------------------------------
// Pass 1: invert the scatter. table[slot] = token id writing that slot, or -1.
// ---------------------------------------------------------------------------
__global__ __launch_bounds__(256) void kv_init_table(int* __restrict__ table) {
    table[(size_t)blockIdx.x * 256 + threadIdx.x] = -1;
}

__global__ __launch_bounds__(256) void kv_fill_table(const int* __restrict__ bidx,
                                                     const int* __restrict__ boff,
                                                     int* __restrict__ table) {
    int t = blockIdx.x * 256 + threadIdx.x;
    if (t < NUM_TOKENS_) {
        int slot = bidx[t] * BLOCK_SIZE_ + boff[t];   // unique slots -> race-free
        table[slot] = t;
    }
}

// ---------------------------------------------------------------------------
// Pass 2 (primary): one 256-thread block per slot, streamed through the CDNA5
// async global<->LDS engine. The slot's source row is either the scattered
// input token or the original cache row (uniform per block -> scalar branch).
//
// Lane i moves bytes [i*16, i*16+16) of the 4KB row:
//   global_load_async_to_lds_b128   copies 16 bytes from src at byte i*16 to LDS byte i*16   (GVS: saddr+voff)
//   s_wait_asynccnt 0                                          (own wave only)
//   global_store_async_from_lds_b128 copies 16 bytes from LDS byte i*16 to out at byte i*16
// The 4KB staging tile is dynamic LDS (base offset 0), so the LDS address and
// the global byte offset are the SAME value -> one VGPR feeds both operands.
// Lanes touch disjoint LDS bytes, so no barriers are required; data never
// transits VGPRs. Tracked by ASYNCcnt (a final s_wait_asynccnt 0 covers
// the trailing stores).
// ---------------------------------------------------------------------------
__global__ __launch_bounds__(256) void kv_merge_async(const float* __restrict__ input,
                                                      const float* __restrict__ cache,
                                                      const int* __restrict__ table,
                                                      float* __restrict__ out) {
    extern __shared__ char kv_smem[];                 // 4KB dynamic LDS, base offset 0
    (void)kv_smem;

    const size_t slot = blockIdx.x;
    const int tok = table[slot];                      // uniform -> scalar load
    const float* src = (tok >= 0)
        ? (input + (size_t)tok * SLOT_ELEMS)
        : (cache + slot * SLOT_ELEMS);
    float* dst = out + slot * SLOT_ELEMS;

    unsigned off = threadIdx.x * 16u;                 // LDS addr == global byte offset

    asm volatile("global_load_async_to_lds_b128 %0, %1, %2 offset:0"
                 :: "v"(off), "v"(off), "s"(src)
                 : "memory");
    asm volatile("s_wait_asynccnt 0x0" ::: "memory");
    asm volatile("global_store_async_from_lds_b128 %0, %1, %2 offset:0"
                 :: "v"(off), "v"(off), "s"(dst)
                 : "memory");
    asm volatile("s_wait_asynccnt 0x0" ::: "memory");
}

// ---------------------------------------------------------------------------
// Pass 2 (alternate, VGPR path): same structure with NT B128 through VGPRs.
// Kept for the no-workspace fallback building block and as a known-good path.
// ---------------------------------------------------------------------------
__global__ __launch_bounds__(256) void kv_copy_cache(const v4f* __restrict__ cache,
                                                     v4f* __restrict__ out) {
    size_t i = (size_t)blockIdx.x * 256 + threadIdx.x;
    v4f v = __builtin_nontemporal_load(cache + i);
    __builtin_nontemporal_store(v, out + i);
}

__global__ __launch_bounds__(SLOT_VEC4) void kv_scatter(const v4f* __restrict__ input,
                                                        const int* __restrict__ bidx,
                                                        const int* __restrict__ boff,
                                                        v4f* __restrict__ out) {
    const size_t t = blockIdx.x;                                 // one block per token
    const size_t slot = (size_t)bidx[t] * BLOCK_SIZE_ + boff[t]; // uniform -> scalar
    v4f v = __builtin_nontemporal_load(input + t * SLOT_VEC4 + threadIdx.x);
    __builtin_nontemporal_store(v, out + slot * SLOT_VEC4 + threadIdx.x);
}

extern "C" void kernel_launch(void* const* d_in, const int* in_sizes, int n_in,
                              void* d_out, int out_size, void* d_ws, size_t ws_size,
                              hipStream_t stream) {
    const float* input = (const float*)d_in[0];   // [16384, 8, 128] f32
    const float* cache = (const float*)d_in[1];   // [1024, 128, 8, 128] f32
    const int*   bidx  = (const int*)d_in[2];     // [16384]
    const int*   boff  = (const int*)d_in[3];     // [16384]
    float*       out   = (float*)d_out;           // full updated cache

    (void)in_sizes; (void)n_in; (void)out_size;

    if (ws_size >= (size_t)NUM_SLOTS * sizeof(int)) {
        // Optimal single-pass path: skip reading cache rows that get overwritten.
        int* table = (int*)d_ws;
        kv_init_table<<<NUM_SLOTS / 256, 256, 0, stream>>>(table);
        kv_fill_table<<<(NUM_TOKENS_ + 255) / 256, 256, 0, stream>>>(bidx, boff, table);
        kv_merge_async<<<NUM_SLOTS, 256, SLOT_BYTES, stream>>>(input, cache, table, out);
    } else {
        // Fallback: copy everything, then scatter tokens (stream-ordered).
        kv_copy_cache<<<(NUM_SLOTS * SLOT_VEC4) / 256, 256, 0, stream>>>(
            (const v4f*)cache, (v4f*)out);
        kv_scatter<<<NUM_TOKENS_, SLOT_VEC4, 0, stream>>>(
            (const v4f*)input, bidx, boff, (v4f*)out);
    }
}